// AttentionDTI_41704132444949
// MI455X (gfx1250) — compile-verified
//
#include <hip/hip_runtime.h>
#include <hip/hip_bf16.h>

typedef __attribute__((ext_vector_type(2))) float v2f;
typedef __attribute__((ext_vector_type(8))) float v8f;

#define B_SZ   4
#define DL     150
#define PL     979
#define CH     160
#define PROTL  1000
#define EMB    64

// ---------------- workspace layout (float offsets) ----------------
#define OFF_PE     0                         // 4*64*1000   = 256000
#define OFF_H1     (OFF_PE + 256000)         // 4*40*997    = 159520
#define OFF_H2     (OFF_H1 + 159520)         // 4*80*990    = 316800
#define OFF_PCONV  (OFF_H2 + 316800)         // 4*160*979   = 626560
#define OFF_DATT   (OFF_PCONV + 626560)      // 4*150*160   = 96000
#define OFF_PATT   (OFF_DATT + 96000)        // 4*979*160   = 626560
#define OFF_CMEAN  (OFF_PATT + 626560)       // 96000
#define OFF_PMEAN  (OFF_CMEAN + 96000)       // 626560
#define OFF_CGATE  (OFF_PMEAN + 626560)      // 96000
#define OFF_PGATE  (OFF_CGATE + 96000)       // 626560
#define OFF_PAIR   (OFF_PGATE + 626560)      // 4*320 = 1280
#define OFF_F1     (OFF_PAIR + 1280)         // 4*1024
#define OFF_F2     (OFF_F1 + 4096)           // 4*1024
#define OFF_F3     (OFF_F2 + 4096)           // 4*512

__device__ __forceinline__ int imin(int a, int b) { return a < b ? a : b; }

// ---------------- templated WMMA f32 GEMM, 2x2 register tiling ----------------
// One wave computes a 32x32 tile of C = act( A(MxK) * B(KxN) + bias ).
// AMODE: 0 A row-major (lda), 1 A transposed: A(m,k)=Ap[k*lda+m]
// BMODE: 0 B row-major (ldb), 1 B transposed: B(k,n)=Bp[n*ldb+k],
//        2 im2col view: B(k,n)=Bp[(k/KSZ)*Lin + n + k%KSZ]
// BIAS:  1 per-col bias[n], 2 per-row bias[m]
// ACT:   0 none, 1 relu, 2 sigmoid, 3 leaky(0.01)
// OOB A-rows / B-cols are CLAMPED (in-bounds reads); their garbage only
// reaches C rows/cols that the masked store never writes.
template <int AMODE, int BMODE, int BIAS, int ACT, int KSZ>
__global__ __launch_bounds__(32) void gemm_wmma_f32(
        const float* __restrict__ A, const float* __restrict__ Bp,
        const float* __restrict__ bias, float* __restrict__ C,
        int M, int N, int K, int lda, int ldb, int ldc,
        long long sA, long long sB, long long sC, int Lin) {
    const int tileN = blockIdx.x * 32;
    const int tileM = blockIdx.y * 32;
    const int bz    = blockIdx.z;
    const float* Ab = A  + (long long)bz * sA;
    const float* Bb = Bp + (long long)bz * sB;
    float*       Cb = C  + (long long)bz * sC;

    const int lane = threadIdx.x & 31;
    const int half = lane >> 4;      // 0: lanes 0-15, 1: lanes 16-31
    const int l16  = lane & 15;

    // clamped fragment indices (always in-bounds)
    const int rowA0 = imin(tileM + l16,      M - 1);
    const int rowA1 = imin(tileM + 16 + l16, M - 1);
    const int colB0 = imin(tileN + l16,      N - 1);
    const int colB1 = imin(tileN + 16 + l16, N - 1);

    // ---- A pointer setup: one running pointer + constant delta to 2nd row-tile
    const float* aptr;
    long long aInc, aS1, dA;
    if constexpr (AMODE == 0) {
        aptr = Ab + (long long)rowA0 * lda + 2 * half;
        aInc = 4; aS1 = 1;
        dA = (long long)(rowA1 - rowA0) * lda;
    } else {
        aptr = Ab + rowA0 + (long long)(2 * half) * lda;
        aInc = 4LL * lda; aS1 = lda;
        dA = rowA1 - rowA0;
    }

    // ---- B pointer setup: one running pointer/offset + constant delta
    const float* bptr = nullptr;
    long long bInc = 0, bS1 = 0, dB = 0;
    long long offx = 0;          // im2col running offset
    int ko = 0;                  // im2col k % KSZ (always even)
    if constexpr (BMODE == 0) {
        bptr = Bb + (long long)(2 * half) * ldb + colB0;
        bInc = 4LL * ldb; bS1 = ldb;
        dB = colB1 - colB0;
    } else if constexpr (BMODE == 1) {
        bptr = Bb + (long long)colB0 * ldb + 2 * half;
        bInc = 4; bS1 = 1;
        dB = (long long)(colB1 - colB0) * ldb;
    } else {
        ko = 2 * half;           // KSZ >= 4 so ci = 0 at start
        offx = colB0 + ko;
        dB = colB1 - colB0;
    }

    v8f acc00 = {}, acc01 = {}, acc10 = {}, acc11 = {};
    #pragma unroll 2
    for (int k0 = 0; k0 < K; k0 += 4) {
        v2f a0, a1, b0, b1;
        a0.x = aptr[0];        a0.y = aptr[aS1];
        a1.x = aptr[dA];       a1.y = aptr[dA + aS1];
        if constexpr (BMODE == 2) {
            b0.x = Bb[offx];        b0.y = Bb[offx + 1];      // ko even -> no wrap on +1
            b1.x = Bb[offx + dB];   b1.y = Bb[offx + dB + 1];
        } else {
            b0.x = bptr[0];        b0.y = bptr[bS1];
            b1.x = bptr[dB];       b1.y = bptr[dB + bS1];
        }
        acc00 = __builtin_amdgcn_wmma_f32_16x16x4_f32(false, a0, false, b0,
                                                      (short)0, acc00, false, false);
        acc01 = __builtin_amdgcn_wmma_f32_16x16x4_f32(false, a0, false, b1,
                                                      (short)0, acc01, false, false);
        acc10 = __builtin_amdgcn_wmma_f32_16x16x4_f32(false, a1, false, b0,
                                                      (short)0, acc10, false, false);
        acc11 = __builtin_amdgcn_wmma_f32_16x16x4_f32(false, a1, false, b1,
                                                      (short)0, acc11, false, false);
        aptr += aInc;
        if constexpr (BMODE == 2) {
            const bool wrap = (ko + 4 >= KSZ);          // step 4 <= KSZ: <=1 wrap
            offx += wrap ? (long long)(4 - KSZ + Lin) : 4LL;
            ko    = wrap ? (ko + 4 - KSZ) : (ko + 4);
        } else {
            bptr += bInc;
        }
    }

    // ---- epilogue: 4 sub-tiles, masked stores ----
    const int colA = tileN + l16;        // sub-tile col for *0 accumulators
    const int colBt = tileN + 16 + l16;  // sub-tile col for *1 accumulators
    #pragma unroll
    for (int r = 0; r < 8; ++r) {
        const int row0 = tileM + r + 8 * half;       // rows of acc0x
        const int row1 = tileM + 16 + r + 8 * half;  // rows of acc1x
        float v00 = acc00[r], v01 = acc01[r], v10 = acc10[r], v11 = acc11[r];
        if constexpr (BIAS == 1) {
            if (colA  < N) { v00 += bias[colA];  v10 += bias[colA];  }
            if (colBt < N) { v01 += bias[colBt]; v11 += bias[colBt]; }
        } else if constexpr (BIAS == 2) {
            if (row0 < M) { v00 += bias[row0]; v01 += bias[row0]; }
            if (row1 < M) { v10 += bias[row1]; v11 += bias[row1]; }
        }
        if constexpr (ACT == 1) {
            v00 = fmaxf(v00, 0.0f); v01 = fmaxf(v01, 0.0f);
            v10 = fmaxf(v10, 0.0f); v11 = fmaxf(v11, 0.0f);
        } else if constexpr (ACT == 2) {
            v00 = 1.0f / (1.0f + __expf(-v00)); v01 = 1.0f / (1.0f + __expf(-v01));
            v10 = 1.0f / (1.0f + __expf(-v10)); v11 = 1.0f / (1.0f + __expf(-v11));
        } else if constexpr (ACT == 3) {
            v00 = (v00 > 0.0f) ? v00 : 0.01f * v00;
            v01 = (v01 > 0.0f) ? v01 : 0.01f * v01;
            v10 = (v10 > 0.0f) ? v10 : 0.01f * v10;
            v11 = (v11 > 0.0f) ? v11 : 0.01f * v11;
        }
        if (row0 < M) {
            if (colA  < N) Cb[(long long)row0 * ldc + colA]  = v00;
            if (colBt < N) Cb[(long long)row0 * ldc + colBt] = v01;
        }
        if (row1 < M) {
            if (colA  < N) Cb[(long long)row1 * ldc + colA]  = v10;
            if (colBt < N) Cb[(long long)row1 * ldc + colBt] = v11;
        }
    }
}

// ---------------- embedding gather ----------------
__global__ void embed_prot(const int* __restrict__ protein,
                           const float* __restrict__ emb, float* __restrict__ pe) {
    long long idx = (long long)blockIdx.x * 256 + threadIdx.x; // B*64*1000
    if (idx >= (long long)B_SZ * EMB * PROTL) return;
    int l = (int)(idx % PROTL);
    int c = (int)((idx / PROTL) % EMB);
    int b = (int)(idx / ((long long)EMB * PROTL));
    int tok = protein[b * PROTL + l];
    pe[idx] = emb[tok * EMB + c];
}

// ---------------- masking ----------------
__global__ void mask_drug(float* __restrict__ da, const int* __restrict__ num_atoms) {
    long long idx = (long long)blockIdx.x * 256 + threadIdx.x; // B*150*160
    if (idx >= (long long)B_SZ * DL * CH) return;
    int b = (int)(idx / (DL * CH));
    int i = (int)((idx / CH) % DL);
    if (i >= num_atoms[b]) da[idx] = 0.0f;
}

__global__ void mask_prot(float* __restrict__ pa, const int* __restrict__ protein) {
    long long idx = (long long)blockIdx.x * 256 + threadIdx.x; // B*979*160
    if (idx >= (long long)B_SZ * PL * CH) return;
    int b = (int)(idx / (PL * CH));
    int j = (int)((idx / CH) % PL);
    if (protein[b * PROTL + j] == 0) pa[idx] = 0.0f;
}

// ---------------- pairwise relu outer-sum means ----------------
// one block per (b, c): Cmean[b,i,c] = mean_j relu(u_i+v_j); Pmean[b,j,c] = mean_i relu(...)
__global__ void attn_pair(const float* __restrict__ da, const float* __restrict__ pa,
                          float* __restrict__ Cmean, float* __restrict__ Pmean) {
    __shared__ float u[DL];
    __shared__ float v[PL];
    __shared__ float red[128];
    const int bc = blockIdx.x;
    const int b = bc / CH;
    const int c = bc % CH;
    const int t = threadIdx.x;

    for (int i = t; i < DL; i += 128) u[i] = da[((long long)b * DL + i) * CH + c];
    for (int j = t; j < PL; j += 128) v[j] = pa[((long long)b * PL + j) * CH + c];
    __syncthreads();

    float cs[8];
    #pragma unroll
    for (int q = 0; q < 8; ++q) cs[q] = 0.0f;

    for (int i = 0; i < DL; ++i) {
        const float ui = u[i];
        float part = 0.0f;
        #pragma unroll
        for (int q = 0; q < 8; ++q) {
            int j = t + q * 128;
            if (j < PL) {
                float r = fmaxf(ui + v[j], 0.0f);
                part += r;
                cs[q] += r;
            }
        }
        red[t] = part;
        __syncthreads();
        for (int s = 64; s > 0; s >>= 1) {
            if (t < s) red[t] += red[t + s];
            __syncthreads();
        }
        if (t == 0) Cmean[((long long)b * DL + i) * CH + c] = red[0] * (1.0f / PL);
        __syncthreads();
    }
    #pragma unroll
    for (int q = 0; q < 8; ++q) {
        int j = t + q * 128;
        if (j < PL) Pmean[((long long)b * PL + j) * CH + c] = cs[q] * (1.0f / DL);
    }
}

// ---------------- gated max pooling ----------------
__global__ void pool_drug(const float* __restrict__ drug, const float* __restrict__ gate,
                          float* __restrict__ pair) {
    int idx = blockIdx.x * 64 + threadIdx.x; // B*160
    if (idx >= B_SZ * CH) return;
    int b = idx / CH, c = idx % CH;
    float m = -3.4e38f;
    for (int l = 0; l < DL; ++l) {
        long long o = ((long long)b * DL + l) * CH + c;
        m = fmaxf(m, drug[o] * (0.5f + gate[o]));
    }
    pair[b * (2 * CH) + c] = m;
}

__global__ void pool_prot(const float* __restrict__ pconv, const float* __restrict__ gate,
                          float* __restrict__ pair) {
    int idx = blockIdx.x * 64 + threadIdx.x; // B*160
    if (idx >= B_SZ * CH) return;
    int b = idx / CH, c = idx % CH;
    float m = -3.4e38f;
    for (int j = 0; j < PL; ++j) {
        float x = pconv[((long long)b * CH + c) * PL + j];
        float g = gate[((long long)b * PL + j) * CH + c];
        m = fmaxf(m, x * (0.5f + g));
    }
    pair[b * (2 * CH) + CH + c] = m;
}

// ---------------- host-side helper ----------------
template <int AMODE, int BMODE, int BIAS, int ACT, int KSZ>
static inline void gemm(hipStream_t s, const float* A, const float* B, const float* bias,
                        float* C, int M, int N, int K, int lda, int ldb, int ldc,
                        long long sA, long long sB, long long sC, int Z, int Lin) {
    dim3 grid((N + 31) / 32, (M + 31) / 32, Z);
    gemm_wmma_f32<AMODE, BMODE, BIAS, ACT, KSZ><<<grid, 32, 0, s>>>(
        A, B, bias, C, M, N, K, lda, ldb, ldc, sA, sB, sC, Lin);
}

extern "C" void kernel_launch(void* const* d_in, const int* in_sizes, int n_in,
                              void* d_out, int out_size, void* d_ws, size_t ws_size,
                              hipStream_t stream) {
    const float* drug      = (const float*)d_in[0];
    const int*   num_atoms = (const int*)d_in[1];
    const int*   protein   = (const int*)d_in[2];
    const float* emb       = (const float*)d_in[3];
    const float* w1 = (const float*)d_in[4];  const float* b1 = (const float*)d_in[5];
    const float* w2 = (const float*)d_in[6];  const float* b2 = (const float*)d_in[7];
    const float* w3 = (const float*)d_in[8];  const float* b3 = (const float*)d_in[9];
    const float* Wd = (const float*)d_in[10]; const float* bd = (const float*)d_in[11];
    const float* Wp = (const float*)d_in[12]; const float* bp = (const float*)d_in[13];
    const float* Wa = (const float*)d_in[14]; const float* ba = (const float*)d_in[15];
    const float* W1 = (const float*)d_in[16]; const float* bf1 = (const float*)d_in[17];
    const float* W2 = (const float*)d_in[18]; const float* bf2 = (const float*)d_in[19];
    const float* W3 = (const float*)d_in[20]; const float* bf3 = (const float*)d_in[21];
    const float* Wo = (const float*)d_in[22]; const float* bo = (const float*)d_in[23];

    float* ws    = (float*)d_ws;
    float* pe    = ws + OFF_PE;
    float* h1    = ws + OFF_H1;
    float* h2    = ws + OFF_H2;
    float* pconv = ws + OFF_PCONV;
    float* datt  = ws + OFF_DATT;
    float* patt  = ws + OFF_PATT;
    float* cmean = ws + OFF_CMEAN;
    float* pmean = ws + OFF_PMEAN;
    float* cgate = ws + OFF_CGATE;
    float* pgate = ws + OFF_PGATE;
    float* pair  = ws + OFF_PAIR;
    float* f1    = ws + OFF_F1;
    float* f2    = ws + OFF_F2;
    float* f3    = ws + OFF_F3;
    float* outp  = (float*)d_out;

    // 1) protein embedding gather: pe (B,64,1000)
    embed_prot<<<(B_SZ * EMB * PROTL + 255) / 256, 256, 0, stream>>>(protein, emb, pe);

    // 2) conv1: (B,40,997) = relu(w1 (40 x 256) @ im2col(pe)), KSZ=4
    gemm<0, 2, 2, 1, 4>(stream, w1, pe, b1, h1, 40, 997, 256, 256, 0, 997,
                        0, (long long)EMB * PROTL, 40LL * 997, B_SZ, PROTL);
    // 3) conv2: (B,80,990) = relu(w2 (80 x 320) @ im2col(h1)), KSZ=8
    gemm<0, 2, 2, 1, 8>(stream, w2, h1, b2, h2, 80, 990, 320, 320, 0, 990,
                        0, 40LL * 997, 80LL * 990, B_SZ, 997);
    // 4) conv3: (B,160,979) = relu(w3 (160 x 960) @ im2col(h2)), KSZ=12
    gemm<0, 2, 2, 1, 12>(stream, w3, h2, b3, pconv, 160, 979, 960, 960, 0, 979,
                         0, 80LL * 990, 160LL * 979, B_SZ, 990);

    // 5) drug_att = drug @ Wd^T + bd          (B,150,160)
    gemm<0, 1, 1, 0, 1>(stream, drug, Wd, bd, datt, DL, CH, CH, CH, CH, CH,
                        (long long)DL * CH, 0, (long long)DL * CH, B_SZ, 0);
    // 6) protein_att = pconv^T @ Wp^T + bp    (B,979,160)
    gemm<1, 1, 1, 0, 1>(stream, pconv, Wp, bp, patt, PL, CH, CH, PL, CH, CH,
                        (long long)CH * PL, 0, (long long)PL * CH, B_SZ, 0);

    // 7/8) masks
    mask_drug<<<(B_SZ * DL * CH + 255) / 256, 256, 0, stream>>>(datt, num_atoms);
    mask_prot<<<(B_SZ * PL * CH + 255) / 256, 256, 0, stream>>>(patt, protein);

    // 9) pairwise relu outer-sum means
    attn_pair<<<B_SZ * CH, 128, 0, stream>>>(datt, patt, cmean, pmean);

    // 10/11) sigmoid attention gates
    gemm<0, 1, 1, 2, 1>(stream, cmean, Wa, ba, cgate, DL, CH, CH, CH, CH, CH,
                        (long long)DL * CH, 0, (long long)DL * CH, B_SZ, 0);
    gemm<0, 1, 1, 2, 1>(stream, pmean, Wa, ba, pgate, PL, CH, CH, CH, CH, CH,
                        (long long)PL * CH, 0, (long long)PL * CH, B_SZ, 0);

    // 12) gated max pools -> pair (B,320)
    pool_drug<<<(B_SZ * CH + 63) / 64, 64, 0, stream>>>(drug, cgate, pair);
    pool_prot<<<(B_SZ * CH + 63) / 64, 64, 0, stream>>>(pconv, pgate, pair);

    // 13-16) MLP head (batch rows contiguous -> M=4)
    gemm<0, 1, 1, 3, 1>(stream, pair, W1, bf1, f1, B_SZ, 1024, 2 * CH,
                        2 * CH, 2 * CH, 1024, 0, 0, 0, 1, 0);
    gemm<0, 1, 1, 3, 1>(stream, f1, W2, bf2, f2, B_SZ, 1024, 1024,
                        1024, 1024, 1024, 0, 0, 0, 1, 0);
    gemm<0, 1, 1, 3, 1>(stream, f2, W3, bf3, f3, B_SZ, 512, 1024,
                        1024, 1024, 512, 0, 0, 0, 1, 0);
    gemm<0, 1, 1, 0, 1>(stream, f3, Wo, bo, outp, B_SZ, 2, 512,
                        512, 512, 2, 0, 0, 0, 1, 0);
}